// GCN_Model_18167711662671
// MI455X (gfx1250) — compile-verified
//
#include <hip/hip_runtime.h>

#define HID 64
#define H4  16   // HID / 4

typedef __attribute__((ext_vector_type(2))) float v2f;
typedef __attribute__((ext_vector_type(8))) float v8f;

// ---------------------------------------------------------------------------
// Degree / normalization kernels
// ---------------------------------------------------------------------------
__global__ void k_deg_init(float* __restrict__ deg, int n) {
    int i = blockIdx.x * blockDim.x + threadIdx.x;
    if (i < n) deg[i] = 1.0f;                 // self-loop contributes 1
}

__global__ void k_deg_scatter(const int* __restrict__ dst, float* __restrict__ deg, int e) {
    int i = blockIdx.x * blockDim.x + threadIdx.x;
    if (i < e) atomicAdd(&deg[dst[i]], 1.0f);
}

__global__ void k_rsqrt(float* __restrict__ deg, int n) {
    int i = blockIdx.x * blockDim.x + threadIdx.x;
    if (i < n) deg[i] = rsqrtf(deg[i]);       // deg >= 1 always (self loops)
}

__global__ void k_norm(const int* __restrict__ src, const int* __restrict__ dst,
                       const float* __restrict__ dis, float* __restrict__ norm, int e) {
    int i = blockIdx.x * blockDim.x + threadIdx.x;
    if (i < e) norm[i] = dis[src[i]] * dis[dst[i]];
}

// ---------------------------------------------------------------------------
// WMMA GEMM:  C[M x NC] = (RELU ? relu(A) : A)[M x K] * B[K x NC] + bias
// One wave32 computes one 16x16 tile of C using V_WMMA_F32_16X16X4_F32.
// M must be a multiple of 16 (N=100000 = 6250*16). NC masked (OUT=40).
// Fragment layouts per CDNA5 ISA 7.12.2:
//   A 16x4 f32 : lanes 0-15 row M=l, v0=K+0, v1=K+1 ; lanes 16-31 v0=K+2, v1=K+3
//   B 4x16 f32 : lanes 0-15 col N=l, v0=K+0, v1=K+1 ; lanes 16-31 v0=K+2, v1=K+3
//   C 16x16    : vgpr i -> row (i + 8*half), col = l
// K, NC, RELU are compile-time: no cndmask in the inner loop, folded strides.
// ---------------------------------------------------------------------------
template <int K, int NC, int RELU>
__global__ __launch_bounds__(256)
void k_gemm(const float* __restrict__ A, const float* __restrict__ B,
            const float* __restrict__ bias, float* __restrict__ C, int M) {
    constexpr int tilesN = (NC + 15) / 16;
    int wave = blockIdx.x * (blockDim.x >> 5) + (threadIdx.x >> 5);
    int lane = threadIdx.x & 31;
    int tiles = (M >> 4) * tilesN;
    if (wave >= tiles) return;                 // wave-uniform exit (EXEC all-1 inside)

    int tm = wave / tilesN;
    int tn = wave - tm * tilesN;
    int hf = lane >> 4;                        // half-wave select
    int l  = lane & 15;
    int col  = tn * 16 + l;
    int colC = (col < NC) ? col : (NC - 1);    // clamp for safe loads (masked tile)

    const v2f*   Ap = (const v2f*)(A + (size_t)(tm * 16 + l) * K + 2 * hf); // 8B aligned
    const float* Bp = B + (size_t)(2 * hf) * NC + colC;

    v8f acc = {};
#pragma unroll 8
    for (int k0 = 0; k0 < K; k0 += 4) {
        v2f a = Ap[k0 >> 1];                   // A[row][k0+2h], A[row][k0+2h+1]
        if (RELU) { a.x = fmaxf(a.x, 0.0f); a.y = fmaxf(a.y, 0.0f); }
        v2f b;
        b.x = Bp[(size_t)k0 * NC];             // B[k0+2h  ][col]
        b.y = Bp[(size_t)(k0 + 1) * NC];       // B[k0+2h+1][col]
        acc = __builtin_amdgcn_wmma_f32_16x16x4_f32(
            /*neg_a=*/false, a, /*neg_b=*/false, b,
            /*c_mod=*/(short)0, acc, /*reuse_a=*/false, /*reuse_b=*/false);
    }

    if (col < NC) {
        float bb = bias[col];
        float* Crow = C + (size_t)(tm * 16 + 8 * hf) * NC + col;
#pragma unroll
        for (int i = 0; i < 8; ++i)
            Crow[(size_t)i * NC] = acc[i] + bb;
    }
}

// ---------------------------------------------------------------------------
// Aggregation: out[i] = tmp[i]*dis[i]^2  (self loop, full overwrite = init)
//              out[dst] += tmp[src]*norm (edge scatter, L2-resident atomics)
// ---------------------------------------------------------------------------
__global__ void k_selfinit(const float* __restrict__ t, const float* __restrict__ dis,
                           float* __restrict__ out, int n) {
    int gid = blockIdx.x * blockDim.x + threadIdx.x;
    if (gid >= n * H4) return;
    int i = gid >> 4;
    float s = dis[i];
    float k = s * s;
    float4 v = ((const float4*)t)[gid];
    float4 r; r.x = v.x * k; r.y = v.y * k; r.z = v.z * k; r.w = v.w * k;
    ((float4*)out)[gid] = r;
}

__global__ void k_scatter(const float* __restrict__ t, const int* __restrict__ src,
                          const int* __restrict__ dst, const float* __restrict__ norm,
                          float* out, int e) {
    int gid = blockIdx.x * blockDim.x + threadIdx.x;
    if (gid >= e * H4) return;
    int ed = gid >> 4;
    int c  = gid & 15;
    int s = src[ed];
    int d = dst[ed];
    float w = norm[ed];
    float4 v = ((const float4*)(t + (size_t)s * HID))[c];   // coalesced 256B/edge
    float* o = out + (size_t)d * HID + (size_t)c * 4;
    atomicAdd(o + 0, v.x * w);     // no-return global_atomic_add_f32, resolves in L2
    atomicAdd(o + 1, v.y * w);
    atomicAdd(o + 2, v.z * w);
    atomicAdd(o + 3, v.w * w);
}

// ---------------------------------------------------------------------------
// Host-side launcher (graph-capture safe: stream-ordered kernels only)
// ---------------------------------------------------------------------------
extern "C" void kernel_launch(void* const* d_in, const int* in_sizes, int n_in,
                              void* d_out, int out_size, void* d_ws, size_t ws_size,
                              hipStream_t stream) {
    (void)n_in; (void)out_size; (void)ws_size;
    const float* x    = (const float*)d_in[0];
    const int*   ei   = (const int*)d_in[1];
    const float* W0   = (const float*)d_in[2];
    const float* b0   = (const float*)d_in[3];
    const float* W1   = (const float*)d_in[4];
    const float* b1   = (const float*)d_in[5];
    const float* W2   = (const float*)d_in[6];
    const float* b2   = (const float*)d_in[7];
    const float* Wout = (const float*)d_in[8];
    const float* bout = (const float*)d_in[9];

    const int FIN = 128, OUTD = 40;
    const int n = in_sizes[0] / FIN;      // 100000
    const int e = in_sizes[1] / 2;        // 1600000
    const int* srcI = ei;                 // edge_index[0]
    const int* dstI = ei + e;             // edge_index[1]

    // Workspace carve-out (256B aligned): dis | norm | tmp | hA | hB  (~84 MB)
    char* w = (char*)d_ws;
    size_t off = 0;
    float* dis  = (float*)(w + off); off += (((size_t)n * 4) + 255) & ~(size_t)255;
    float* norm = (float*)(w + off); off += (((size_t)e * 4) + 255) & ~(size_t)255;
    float* tmp  = (float*)(w + off); off += (((size_t)n * HID * 4) + 255) & ~(size_t)255;
    float* hA   = (float*)(w + off); off += (((size_t)n * HID * 4) + 255) & ~(size_t)255;
    float* hB   = (float*)(w + off); off += (((size_t)n * HID * 4) + 255) & ~(size_t)255;

    const int T   = 256;            // 8 wave32s per block
    const int WPB = T / 32;

    // GCN normalization (computed once, reused by all 3 layers)
    k_deg_init   <<<(n + T - 1) / T, T, 0, stream>>>(dis, n);
    k_deg_scatter<<<(e + T - 1) / T, T, 0, stream>>>(dstI, dis, e);
    k_rsqrt      <<<(n + T - 1) / T, T, 0, stream>>>(dis, n);
    k_norm       <<<(e + T - 1) / T, T, 0, stream>>>(srcI, dstI, dis, norm, e);

    const int tilesH = (n / 16) * (HID / 16);
    const int gGemmH = (tilesH + WPB - 1) / WPB;
    const int gAgg1  = (n * H4 + T - 1) / T;
    const int gAggE  = (e * H4 + T - 1) / T;

    // Layer 0: x[128] @ W0 -> aggregate -> hA
    k_gemm<128, HID, 0><<<gGemmH, T, 0, stream>>>(x, W0, b0, tmp, n);
    k_selfinit<<<gAgg1, T, 0, stream>>>(tmp, dis, hA, n);
    k_scatter <<<gAggE, T, 0, stream>>>(tmp, srcI, dstI, norm, hA, e);

    // Layer 1: relu(hA) @ W1 -> aggregate -> hB
    k_gemm<HID, HID, 1><<<gGemmH, T, 0, stream>>>(hA, W1, b1, tmp, n);
    k_selfinit<<<gAgg1, T, 0, stream>>>(tmp, dis, hB, n);
    k_scatter <<<gAggE, T, 0, stream>>>(tmp, srcI, dstI, norm, hB, e);

    // Layer 2: relu(hB) @ W2 -> aggregate -> hA
    k_gemm<HID, HID, 1><<<gGemmH, T, 0, stream>>>(hB, W2, b2, tmp, n);
    k_selfinit<<<gAgg1, T, 0, stream>>>(tmp, dis, hA, n);
    k_scatter <<<gAggE, T, 0, stream>>>(tmp, srcI, dstI, norm, hA, e);

    // Output linear: relu(hA) @ Wout + bout -> d_out  (NC=40, masked tiles)
    const int tilesO = (n / 16) * ((OUTD + 15) / 16);
    k_gemm<HID, OUTD, 1><<<(tilesO + WPB - 1) / WPB, T, 0, stream>>>(hA, Wout, bout,
                                                                     (float*)d_out, n);
}